// RelationalEncoding_66692252172937
// MI455X (gfx1250) — compile-verified
//
#include <hip/hip_runtime.h>

typedef __attribute__((ext_vector_type(2))) float v2f;
typedef __attribute__((ext_vector_type(8))) float v8f;
typedef __attribute__((ext_vector_type(4))) float f4;

#define BSZ 8
#define SSZ 160
#define DSZ 256
#define MROWS (BSZ * SSZ)             // 1280 rows of x (flattened)
#define NPAIR ((SSZ * (SSZ - 1)) / 2) // 12720
#define PPB 8                         // pairs per expand block (256 thr / 32 lanes)

// ---------------------------------------------------------------------------
// Kernel 1: y[0 .. MROWS*DSZ)        = x @ Wi + bias   (bias folded here)
//           y[MROWS*DSZ .. 2*...)    = x @ Wj
// One wave computes one 16x16 f32 tile with V_WMMA_F32_16X16X4_F32,
// looping K in steps of 4.  Tiles: 2 gemms * (1280/16=80) * (256/16=16) = 2560.
// ---------------------------------------------------------------------------
__global__ __launch_bounds__(256) void relenc_gemm(const float* __restrict__ x,
                                                   const float* __restrict__ W,
                                                   const float* __restrict__ bias,
                                                   float* __restrict__ y) {
  const int wave = blockIdx.x * (blockDim.x >> 5) + (threadIdx.x >> 5);
  const int lane = threadIdx.x & 31;

  const int which = wave / (80 * 16);      // 0 -> Wi, 1 -> Wj
  const int t     = wave % (80 * 16);
  const int tm    = t % 80;                // M tile
  const int tn    = t / 80;                // N tile

  const int mlo = lane & 15;               // M row (A) / N col (B) within tile
  const int hi  = lane >> 4;               // selects K pair {0,1} vs {2,3}

  const float* Wsel = W + (size_t)which * DSZ * DSZ;       // Wi or Wj (stacked rows)
  const float* arow = x + (size_t)(tm * 16 + mlo) * DSZ;   // A row for this lane
  const int    ncol = tn * 16 + mlo;                       // B/D column for this lane

  v8f acc = {};
  for (int k0 = 0; k0 < DSZ; k0 += 4) {
    const int ka = k0 + hi * 2;
    v2f a;                                  // A 16x4: VGPR0=K even, VGPR1=K odd (per half)
    a.x = arow[ka];
    a.y = arow[ka + 1];
    v2f bm;                                 // B 4x16: N striped across lanes
    bm.x = Wsel[(size_t)ka * DSZ + ncol];
    bm.y = Wsel[(size_t)(ka + 1) * DSZ + ncol];
    // 8 args: (neg_a, A, neg_b, B, c_mod, C, reuse_a, reuse_b)
    acc = __builtin_amdgcn_wmma_f32_16x16x4_f32(false, a, false, bm,
                                                (short)0, acc, false, false);
  }

  const float badd = (which == 0) ? bias[ncol] : 0.0f;
  float* yrow = y + (size_t)which * MROWS * DSZ;
#pragma unroll
  for (int v = 0; v < 8; ++v) {
    const int mrow = tm * 16 + v + hi * 8;  // D layout: VGPR v -> row v (lanes 0-15) / v+8
    yrow[(size_t)mrow * DSZ + ncol] = acc[v] + badd;
  }
}

// cumulative pair count before row i for triu(k=1): C(i) = i*(2S-1-i)/2
__device__ __forceinline__ int cfun(int i) { return i * (2 * SSZ - 1 - i) / 2; }

// ---------------------------------------------------------------------------
// Kernel 2: out[b, p, :] = (xa[b,i,:] + bias) + xb[b,j,:]   for p = pair(i,j)
// Dense 1D pair grid: 12720/8 = 1590 blocks in x, batch in y -> 12,720 blocks
// of 256 threads, none wasted.  Each 32-lane group owns one 1KB output row;
// each lane stores 2 x float4 (512B contiguous per wave access).
// (i,j) recovered from flat p via closed-form sqrt inversion + 0/1-step fixup.
// ---------------------------------------------------------------------------
__global__ __launch_bounds__(256) void relenc_expand(const float* __restrict__ y,
                                                     float* __restrict__ out) {
  const int b    = blockIdx.y;
  const int grp  = threadIdx.x >> 5;              // 0..7: pair within block
  const int lane = threadIdx.x & 31;
  const int p    = blockIdx.x * PPB + grp;        // flat pair index, < 12720

  // invert p -> (i, j):  i = floor((2S-1 - sqrt((2S-1)^2 - 8p)) / 2), then fixup
  const float disc = (float)((2 * SSZ - 1) * (2 * SSZ - 1) - 8 * p);
  int i = (int)(((float)(2 * SSZ - 1) - sqrtf(disc)) * 0.5f);
  if (i > 0 && cfun(i) > p) --i;
  if (cfun(i + 1) <= p) ++i;
  const int j = p - cfun(i) + i + 1;

  const f4* xa = (const f4*)(y + (size_t)(b * SSZ + i) * DSZ);
  const f4* xb = (const f4*)(y + (size_t)MROWS * DSZ + (size_t)(b * SSZ + j) * DSZ);
  f4* o = (f4*)(out + ((size_t)b * NPAIR + p) * DSZ);

  // 256 floats = 64 float4 per row; 32 lanes x 2 float4 each
  o[lane]      = xa[lane]      + xb[lane];
  o[lane + 32] = xa[lane + 32] + xb[lane + 32];
}

extern "C" void kernel_launch(void* const* d_in, const int* in_sizes, int n_in,
                              void* d_out, int out_size, void* d_ws, size_t ws_size,
                              hipStream_t stream) {
  const float* x    = (const float*)d_in[0];   // (8,160,256)
  const float* W    = (const float*)d_in[1];   // (512,256)
  const float* bias = (const float*)d_in[2];   // (256,)
  float* out = (float*)d_out;                  // (8,12720,256)
  float* y   = (float*)d_ws;                   // 2 * 1280 * 256 f32 = 2.62 MB

  // 2560 tiles, 8 waves (256 threads) per block -> 320 blocks
  relenc_gemm<<<dim3(320), 256, 0, stream>>>(x, W, bias, y);
  // 12720 pairs / 8 per block = 1590 blocks per batch, 8 batches
  relenc_expand<<<dim3(NPAIR / PPB, BSZ), 256, 0, stream>>>(y, out);
}